// RegionProposalNetwork_27204322853177
// MI455X (gfx1250) — compile-verified
//
#include <hip/hip_runtime.h>
#include <math.h>

typedef float v2f __attribute__((ext_vector_type(2)));
typedef float v8f __attribute__((ext_vector_type(8)));
typedef unsigned int v4u __attribute__((ext_vector_type(4)));
typedef int v8i __attribute__((ext_vector_type(8)));
typedef int v4i __attribute__((ext_vector_type(4)));

#define IMG_N   8
#define CIN     256
#define HH      100
#define WW      100
#define AN      9
#define HW      (HH*WW)
#define HWA     (HW*AN)
#define PRE_NMS 2000
#define POST_NMS 1000
#define NMS_THRESH 0.7f
#define MIN_SIZE 1e-3f
#define NEGV    (-1e9f)
#define BBOX_CLIP 4.135166556742356f   /* log(1000/16) */

#if __has_builtin(__builtin_amdgcn_tensor_load_to_lds)
#define HAVE_TDM 1
#else
#define HAVE_TDM 0
#endif

// ---------------------------------------------------------------------------
// Kernel 0: repack conv weights OIHW -> [j][c_in][c_out] so each WMMA A-tile
// is 16 lines x 64 contiguous floats (TDM-friendly, 128b-load friendly).
// ---------------------------------------------------------------------------
__global__ __launch_bounds__(256) void repack_w(
    const float* __restrict__ wgt, float* __restrict__ wp)
{
  const int id = blockIdx.x * blockDim.x + threadIdx.x;  // 9*256*256
  if (id >= 9 * CIN * CIN) return;
  const int j   = id >> 16;          // /65536
  const int rem = id & 0xFFFF;
  const int c   = rem >> 8;
  const int o   = rem & 255;
  wp[id] = wgt[(o * CIN + c) * 9 + j];
}

// ---------------------------------------------------------------------------
// Kernel 1: 3x3 conv (SAME) + bias + ReLU as implicit GEMM via f32 WMMA.
// Tile: 64 c_out (M) x 64 spatial (N), K=2304 as j(0..8) x 16-ch blocks.
// 8 waves in 4(M) x 2(N); each wave: two 16x16 f32 accumulators sharing one
// A fragment -> 8 v_wmma per K-slab per wave.
// A operand staged by the Tensor Data Mover (wave 0 issues TENSOR_LOAD_TO_LDS,
// s_wait_tensorcnt before the barrier); B operand staged manually (halo).
// ---------------------------------------------------------------------------
__global__ __launch_bounds__(256) void conv3_wmma(
    const float* __restrict__ feats, const float* __restrict__ wp,
    const float* __restrict__ bias, float* __restrict__ t)
{
  __shared__ float As[16][64];   // weights: [k-within-block][c_out]
  __shared__ float Bs[16][64];   // patches: [k-within-block][spatial]

  const int tid  = threadIdx.x;
  const int lane = tid & 31;
  const int wave = tid >> 5;
  const int wm   = wave & 3;     // M sub-tile (c_out, 16 wide)
  const int wn   = wave >> 2;    // N range (spatial, 32 wide -> 2 subtiles)

  const int sb  = blockIdx.x;        // 0..1599 spatial tiles
  const int img = sb / 200;
  const int rem = sb - img * 200;
  const int h   = rem >> 1;          // image row
  const int w0  = (rem & 1) << 6;    // 0 or 64
  const int m0  = blockIdx.y << 6;   // c_out block base (0,64,128,192)

  v8f acc0 = {0.f, 0.f, 0.f, 0.f, 0.f, 0.f, 0.f, 0.f};
  v8f acc1 = {0.f, 0.f, 0.f, 0.f, 0.f, 0.f, 0.f, 0.f};

#if HAVE_TDM
  const unsigned ldsA = (unsigned)(unsigned long long)(&As[0][0]);
#endif

  for (int kb = 0; kb < 144; ++kb) {
    const int j  = kb >> 4;              // kernel position 0..8
    const int c0 = (kb & 15) << 4;       // input-channel block base
    const int kh = j / 3 - 1;
    const int kw = j - (j / 3) * 3 - 1;
    const int hh = h + kh;

#if HAVE_TDM
    // Stage A via Tensor Data Mover: 2D tile, 16 lines x 64 floats,
    // line stride 256 floats, from packed weights. Issued by wave 0 only
    // (TDM ignores EXEC, so it must be branched around, not masked).
    if (wave == 0) {
      const unsigned long long ga =
          (unsigned long long)(const void*)(wp + ((size_t)(j * CIN + c0) * CIN + m0));
      v4u g0;
      g0.x = 0x1u;                                   // count=1 (valid D#)
      g0.y = ldsA;                                   // lds_addr (bytes)
      g0.z = (unsigned)(ga & 0xFFFFFFFFu);           // global_addr[31:0]
      g0.w = (unsigned)((ga >> 32) & 0x1FFFFFFu)     // global_addr[56:32]
             | (2u << 30);                           // type=2 ("image")
      v8i g1;
      g1[0] = (int)(2u << 16);       // workgroup_mask=0, data_size=2 (4B)
      g1[1] = 0;                     // atomic_barrier_addr=0, td0[15:0]=0
      g1[2] = 0x10;                  // tensor_dim0=1<<20 (no OOB), td1 lo=0
      g1[3] = 0x10 | (64 << 16);     // tensor_dim1 hi, tile_dim0=64
      g1[4] = 16;                    // tile_dim1=16, tile_dim2=0 (2D)
      g1[5] = 256;                   // tensor_dim0_stride=256 elements
      g1[6] = 0;
      g1[7] = 0;
      v4i gz4 = {0, 0, 0, 0};
      v8i gz8 = {0, 0, 0, 0, 0, 0, 0, 0};
      // 6-arg form (this toolchain): (g0, g1, g2, g3, g4, cpol)
      __builtin_amdgcn_tensor_load_to_lds(g0, g1, gz4, gz4, gz8, 0);
    }
#else
    // Fallback: contiguous float4 staging of A from packed weights.
    {
      const int kk = tid >> 4;           // 0..15
      const int mm = (tid & 15) << 2;    // 0..60 step 4
      const float4 v = *(const float4*)(wp + ((size_t)(j * CIN + c0 + kk) * CIN + m0 + mm));
      As[kk][mm + 0] = v.x; As[kk][mm + 1] = v.y;
      As[kk][mm + 2] = v.z; As[kk][mm + 3] = v.w;
    }
#endif

    // Stage B (input patches): 1024 elems, 4 per thread, contiguous in w.
    for (int e = tid; e < 1024; e += 256) {
      const int kk = e >> 6, nn = e & 63;
      const int wc = w0 + nn + kw;
      float v = 0.f;
      if (hh >= 0 && hh < HH && wc >= 0 && wc < WW) {
        const float* src = &feats[(((img * CIN + c0 + kk) * HH) + hh) * WW + wc];
        v = *src;
        if ((kb & 15) != 15)                 // next K-block: c0 += 16
          __builtin_prefetch(src + 16 * HW, 0, 1);
      }
      Bs[kk][nn] = v;
    }

#if HAVE_TDM
    if (wave == 0) __builtin_amdgcn_s_wait_tensorcnt(0);
#endif
    __syncthreads();

    // 4 k-steps of depth 4. ISA f32 A/B layout: per VGPR, lanes 0-15 hold
    // K=base+0, lanes 16-31 hold K=base+2; second VGPR holds K=+1/+3.
    const int ml = lane & 15;
    #pragma unroll
    for (int ks = 0; ks < 4; ++ks) {
      const int kbase = ks * 4 + ((lane >> 4) << 1);
      v2f a, b0, b1;
      a.x  = As[kbase + 0][wm * 16 + ml];
      a.y  = As[kbase + 1][wm * 16 + ml];
      b0.x = Bs[kbase + 0][wn * 32 + ml];
      b0.y = Bs[kbase + 1][wn * 32 + ml];
      b1.x = Bs[kbase + 0][wn * 32 + 16 + ml];
      b1.y = Bs[kbase + 1][wn * 32 + 16 + ml];
      acc0 = __builtin_amdgcn_wmma_f32_16x16x4_f32(
          false, a, false, b0, (short)0, acc0, false, false);
      acc1 = __builtin_amdgcn_wmma_f32_16x16x4_f32(
          false, a, false, b1, (short)0, acc1, false, false);
    }
    __syncthreads();
  }

  // Epilogue: C layout — VGPR r: lanes 0-15 M=r, lanes 16-31 M=r+8; N=lane&15.
  const int wcolA = w0 + wn * 32 + (lane & 15);
  const int wcolB = wcolA + 16;
  #pragma unroll
  for (int r = 0; r < 8; ++r) {
    const int co = m0 + wm * 16 + r + ((lane >> 4) << 3);
    const float b = bias[co];
    const size_t row = ((size_t)(img * CIN + co) * HH + h) * WW;
    if (wcolA < WW) {
      float v = acc0[r] + b;
      t[row + wcolA] = v > 0.f ? v : 0.f;
    }
    if (wcolB < WW) {
      float v = acc1[r] + b;
      t[row + wcolB] = v > 0.f ? v : 0.f;
    }
  }
}

// ---------------------------------------------------------------------------
// Kernel 2: 1x1 heads (9 obj + 36 reg) + BoxCoder decode + clip.
// ---------------------------------------------------------------------------
__global__ __launch_bounds__(128) void head_decode(
    const float* __restrict__ t,
    const float* __restrict__ cls_w, const float* __restrict__ cls_b,
    const float* __restrict__ bbox_w, const float* __restrict__ bbox_b,
    const float* __restrict__ anchors,
    float* __restrict__ scores, float* __restrict__ boxes)
{
  const int id = blockIdx.x * blockDim.x + threadIdx.x;
  if (id >= IMG_N * HW) return;
  const int img = id / HW;
  const int hw  = id - img * HW;

  float obj[AN];
  float rg[4 * AN];
  #pragma unroll
  for (int a = 0; a < AN; ++a) obj[a] = 0.f;
  #pragma unroll
  for (int r = 0; r < 4 * AN; ++r) rg[r] = 0.f;

  const float* tp = t + (size_t)img * CIN * HW + hw;
  for (int c = 0; c < CIN; ++c) {
    const float x = tp[(size_t)c * HW];
    #pragma unroll
    for (int a = 0; a < AN; ++a) obj[a] = fmaf(x, cls_w[a * CIN + c], obj[a]);
    #pragma unroll
    for (int r = 0; r < 4 * AN; ++r) rg[r] = fmaf(x, bbox_w[r * CIN + c], rg[r]);
  }

  #pragma unroll
  for (int a = 0; a < AN; ++a) {
    const int aidx = hw * AN + a;
    const float* an = anchors + (size_t)aidx * 4;
    const float x1a = an[0], y1a = an[1], x2a = an[2], y2a = an[3];
    const float wa = x2a - x1a, ha = y2a - y1a;
    const float cxa = x1a + 0.5f * wa, cya = y1a + 0.5f * ha;
    const float dx = rg[a * 4 + 0] + bbox_b[a * 4 + 0];
    const float dy = rg[a * 4 + 1] + bbox_b[a * 4 + 1];
    const float dw = fminf(rg[a * 4 + 2] + bbox_b[a * 4 + 2], BBOX_CLIP);
    const float dh = fminf(rg[a * 4 + 3] + bbox_b[a * 4 + 3], BBOX_CLIP);
    const float pcx = dx * wa + cxa, pcy = dy * ha + cya;
    const float pw = expf(dw) * wa, ph = expf(dh) * ha;
    float bx1 = fminf(fmaxf(pcx - 0.5f * pw, 0.f), 800.f);
    float by1 = fminf(fmaxf(pcy - 0.5f * ph, 0.f), 800.f);
    float bx2 = fminf(fmaxf(pcx + 0.5f * pw, 0.f), 800.f);
    float by2 = fminf(fmaxf(pcy + 0.5f * ph, 0.f), 800.f);
    const size_t o = (size_t)img * HWA + aidx;
    boxes[o * 4 + 0] = bx1;
    boxes[o * 4 + 1] = by1;
    boxes[o * 4 + 2] = bx2;
    boxes[o * 4 + 3] = by2;
    scores[o] = obj[a] + cls_b[a];
  }
}

// ---------------------------------------------------------------------------
// Kernel 3: per-image top-2000 (sorted desc, low-index tie-break like top_k).
// ---------------------------------------------------------------------------
__global__ __launch_bounds__(256) void topk_kernel(
    float* __restrict__ scores, const float* __restrict__ boxes,
    float* __restrict__ topS, float* __restrict__ topB)
{
  __shared__ float smax[256];
  __shared__ int   sidx[256];
  const int img = blockIdx.x, tid = threadIdx.x;
  float* S = scores + (size_t)img * HWA;
  const float* B = boxes + (size_t)img * HWA * 4;

  float lmax = -INFINITY;
  int lidx = 0;
  for (int i = tid; i < HWA; i += 256) {
    const float v = S[i];
    if (v > lmax) { lmax = v; lidx = i; }
  }

  for (int k = 0; k < PRE_NMS; ++k) {
    smax[tid] = lmax; sidx[tid] = lidx;
    __syncthreads();
    for (int off = 128; off > 0; off >>= 1) {
      if (tid < off) {
        const float vo = smax[tid + off];
        const int   io = sidx[tid + off];
        if (vo > smax[tid] || (vo == smax[tid] && io < sidx[tid])) {
          smax[tid] = vo; sidx[tid] = io;
        }
      }
      __syncthreads();
    }
    const int gidx = sidx[0];
    if (tid == 0) {
      topS[img * PRE_NMS + k] = smax[0];
      topB[(img * PRE_NMS + k) * 4 + 0] = B[(size_t)gidx * 4 + 0];
      topB[(img * PRE_NMS + k) * 4 + 1] = B[(size_t)gidx * 4 + 1];
      topB[(img * PRE_NMS + k) * 4 + 2] = B[(size_t)gidx * 4 + 2];
      topB[(img * PRE_NMS + k) * 4 + 3] = B[(size_t)gidx * 4 + 3];
    }
    if (tid == (gidx & 255)) {           // owner lane: consume + rescan slice
      S[gidx] = -INFINITY;
      lmax = -INFINITY; lidx = 0;
      for (int i = tid; i < HWA; i += 256) {
        const float v = S[i];
        if (v > lmax) { lmax = v; lidx = i; }
      }
    }
    __syncthreads();
  }
}

// ---------------------------------------------------------------------------
// Kernel 4: per-image greedy NMS in LDS + rank scatter into [1000,5] output.
// ---------------------------------------------------------------------------
__global__ __launch_bounds__(256) void nms_kernel(
    const float* __restrict__ topS, const float* __restrict__ topB,
    float* __restrict__ out)
{
  __shared__ float X1[PRE_NMS], Y1[PRE_NMS], X2[PRE_NMS], Y2[PRE_NMS];
  __shared__ float AR[PRE_NMS], SC[PRE_NMS];
  __shared__ unsigned char SUP[PRE_NMS];
  const int img = blockIdx.x, tid = threadIdx.x;

  for (int i = tid; i < PRE_NMS; i += 256) {
    const float x1 = topB[((size_t)img * PRE_NMS + i) * 4 + 0];
    const float y1 = topB[((size_t)img * PRE_NMS + i) * 4 + 1];
    const float x2 = topB[((size_t)img * PRE_NMS + i) * 4 + 2];
    const float y2 = topB[((size_t)img * PRE_NMS + i) * 4 + 3];
    const bool valid = ((x2 - x1) >= MIN_SIZE) && ((y2 - y1) >= MIN_SIZE);
    X1[i] = x1; Y1[i] = y1; X2[i] = x2; Y2[i] = y2;
    AR[i] = (x2 - x1) * (y2 - y1);
    SC[i] = valid ? topS[(size_t)img * PRE_NMS + i] : NEGV;
    SUP[i] = valid ? 0 : 1;
  }
  for (int i = tid; i < POST_NMS; i += 256) {
    const size_t o = ((size_t)img * POST_NMS + i) * 5;
    out[o + 0] = 0.f; out[o + 1] = 0.f; out[o + 2] = 0.f; out[o + 3] = 0.f;
    out[o + 4] = NEGV;
  }
  __syncthreads();

  for (int i = 0; i < PRE_NMS; ++i) {
    if (!SUP[i]) {
      const float x1 = X1[i], y1 = Y1[i], x2 = X2[i], y2 = Y2[i], ai = AR[i];
      for (int jj = i + 1 + tid; jj < PRE_NMS; jj += 256) {
        if (!SUP[jj]) {
          const float lx = fmaxf(x1, X1[jj]), ly = fmaxf(y1, Y1[jj]);
          const float rx = fminf(x2, X2[jj]), ry = fminf(y2, Y2[jj]);
          const float iw = fmaxf(rx - lx, 0.f), ih = fmaxf(ry - ly, 0.f);
          const float inter = iw * ih;
          const float uni = fmaxf(ai + AR[jj] - inter, 1e-9f);
          if (inter > NMS_THRESH * uni) SUP[jj] = 1;
        }
      }
    }
    __syncthreads();
  }

  if (tid == 0) {
    int rank = 0;
    for (int i = 0; i < PRE_NMS && rank < POST_NMS; ++i) {
      if (!SUP[i]) {
        const size_t o = ((size_t)img * POST_NMS + rank) * 5;
        out[o + 0] = X1[i]; out[o + 1] = Y1[i];
        out[o + 2] = X2[i]; out[o + 3] = Y2[i];
        out[o + 4] = SC[i];
        ++rank;
      }
    }
  }
}

// ---------------------------------------------------------------------------
extern "C" void kernel_launch(void* const* d_in, const int* in_sizes, int n_in,
                              void* d_out, int out_size, void* d_ws, size_t ws_size,
                              hipStream_t stream)
{
  const float* feats   = (const float*)d_in[0];
  const float* conv_w  = (const float*)d_in[1];
  const float* conv_b  = (const float*)d_in[2];
  const float* cls_w   = (const float*)d_in[3];
  const float* cls_b   = (const float*)d_in[4];
  const float* bbox_w  = (const float*)d_in[5];
  const float* bbox_b  = (const float*)d_in[6];
  const float* anchors = (const float*)d_in[7];

  float* ws     = (float*)d_ws;
  float* t      = ws;                    // 20,480,000 f (81.9 MB)
  float* scores = t + 20480000;          //    720,000 f
  float* boxes  = scores + 720000;       //  2,880,000 f
  float* topS   = boxes + 2880000;       //     16,000 f
  float* topB   = topS + 16000;          //     64,000 f
  float* wp     = topB + 64000;          //    589,824 f (packed weights)
  float* out    = (float*)d_out;         // 8*1000*5

  repack_w<<<(9 * CIN * CIN + 255) / 256, 256, 0, stream>>>(conv_w, wp);
  conv3_wmma<<<dim3(1600, 4), 256, 0, stream>>>(feats, wp, conv_b, t);
  head_decode<<<(IMG_N * HW + 127) / 128, 128, 0, stream>>>(
      t, cls_w, cls_b, bbox_w, bbox_b, anchors, scores, boxes);
  topk_kernel<<<IMG_N, 256, 0, stream>>>(scores, boxes, topS, topB);
  nms_kernel<<<IMG_N, 256, 0, stream>>>(topS, topB, out);
}